// GRU_77653008711699
// MI455X (gfx1250) — compile-verified
//
#include <hip/hip_runtime.h>
#include <math.h>
#include <stdint.h>

typedef __attribute__((ext_vector_type(2))) float v2f;
typedef __attribute__((ext_vector_type(8))) float v8f;
typedef __attribute__((ext_vector_type(4))) unsigned int v4u;
typedef __attribute__((ext_vector_type(4))) int v4i;
typedef __attribute__((ext_vector_type(8))) int v8i;

#define GRU_IN   262144
#define GRU_T    30
#define KC       128            // K-chunk per LDS stage
#define XST      132            // padded LDS row stride for x (bank-conflict free)
#define NSTRIPE  8              // striped accumulators to cut atomic contention
#define NCHUNK   (GRU_IN / KC)  // 2048
#define NBLOCK   1024
#define GSTRIDE  (3 * 32 * 16)  // one stripe: [3 gates][32 padded t][16]

// ---- TDM: one 2D (or 1D) tile Global->LDS, descriptor per CDNA5 ISA ch.8 ----
// w0flags = group1 word0: data_size[17:16], pad_enable[20], pad_interval[24:22],
//           pad_amount[31:25]
__device__ __forceinline__ void tdm_load(uint32_t lds_off, const void* gptr,
                                         uint32_t w0flags,
                                         uint32_t td0, uint32_t td1,
                                         uint32_t tile0, uint32_t tile1,
                                         uint64_t stride0)
{
  const uint64_t ga = (uint64_t)gptr;
  v4u g0;
  g0.x = 1u;                                   // count=1 valid descriptor
  g0.y = lds_off;                              // lds_addr (bytes)
  g0.z = (uint32_t)ga;                         // global_addr[31:0]
  g0.w = (uint32_t)(ga >> 32) | 0x80000000u;   // global_addr[56:32] | type=2
  v8i g1;
  g1[0] = (int)w0flags;                                        // wg_mask=0 etc.
  g1[1] = (int)((td0 & 0xFFFFu) << 16);                        // tensor_dim0 lo
  g1[2] = (int)((td0 >> 16) | ((td1 & 0xFFFFu) << 16));        // td0 hi | td1 lo
  g1[3] = (int)((td1 >> 16) | (tile0 << 16));                  // td1 hi | tile0
  g1[4] = (int)tile1;                                          // tile1 | tile2=0
  g1[5] = (int)(uint32_t)stride0;                              // dim0 stride lo
  g1[6] = (int)(uint32_t)(stride0 >> 32);                      // stride hi
  g1[7] = 0;
  v4i z4 = {0, 0, 0, 0};
#if defined(__clang_major__) && (__clang_major__ >= 23)
  v8i z8 = {0, 0, 0, 0, 0, 0, 0, 0};
  __builtin_amdgcn_tensor_load_to_lds(g0, g1, z4, z4, z8, 0);
#else
  __builtin_amdgcn_tensor_load_to_lds(g0, g1, z4, z4, 0);
#endif
}

// ---------------- Phase A: Gx[g][t][j] = sum_k x[t][k] * Wg[k][j] ----------------
__global__ __launch_bounds__(192) void gru_gemm_wmma(
    const float* __restrict__ x,  const float* __restrict__ Wr,
    const float* __restrict__ Wz, const float* __restrict__ Wh,
    float* __restrict__ G)
{
  __shared__ __align__(16) float xs[2][32 * XST];       // 2 x 16.9 KB
  __shared__ __align__(16) float ws[2][3 * KC * 16];    // 2 x 24 KB
  const int tid   = threadIdx.x;
  const int wave  = tid >> 5;
  const int lane  = tid & 31;
  const int mt    = wave & 1;       // t-tile: rows 0-15 / 16-31(padded)
  const int nt    = wave >> 1;      // 0=r, 1=z, 2=h
  const int lhalf = lane >> 4;
  const int lmod  = lane & 15;

  const uint32_t xs_off = (uint32_t)(uintptr_t)&xs[0][0];
  const uint32_t ws_off = (uint32_t)(uintptr_t)&ws[0][0];
  const uint32_t XBUF = 32 * XST * 4;      // bytes per x buffer
  const uint32_t WBUF = 3 * KC * 16 * 4;   // bytes per W buffer
  // x tile: pad_enable, pad_interval=6 (128 DWORDs = one row), pad_amount=3 (4 DWORDs)
  const uint32_t XW0 = (2u << 16) | (1u << 20) | (6u << 22) | (3u << 25);
  const uint32_t WW0 = (2u << 16);         // f32, no padding

  // zero-pad x rows 30,31 of both buffers once (TDM only writes rows < 30)
  if (tid < 4 * (KC / 4)) {
    const int b = tid / (2 * (KC / 4));
    const int r = GRU_T + (tid / (KC / 4)) % 2, c4 = tid % (KC / 4);
    *(float4*)&xs[b][r * XST + c4 * 4] = make_float4(0.f, 0.f, 0.f, 0.f);
  }

  // prologue: DMA first chunk into buffer 0
  const int c0 = blockIdx.x;
  if (wave == 0 && c0 < NCHUNK) {
    const int k0 = c0 * KC;
    tdm_load(xs_off, x + k0, XW0, GRU_IN, GRU_T, KC, GRU_T, GRU_IN);
    tdm_load(ws_off + 0 * KC * 64, Wr + (size_t)k0 * 16, WW0,
             (GRU_IN + 16) * 16, 1, KC * 16, 1, KC * 16);
    tdm_load(ws_off + 1 * KC * 64, Wz + (size_t)k0 * 16, WW0,
             (GRU_IN + 16) * 16, 1, KC * 16, 1, KC * 16);
    tdm_load(ws_off + 2 * KC * 64, Wh + (size_t)k0 * 16, WW0,
             (GRU_IN + 16) * 16, 1, KC * 16, 1, KC * 16);
  }

  v8f acc0 = {};
  v8f acc1 = {};
  int p = 0;

  for (int c = c0; c < NCHUNK; c += gridDim.x) {
    if (wave == 0) __builtin_amdgcn_s_wait_tensorcnt(0);
    __syncthreads();   // chunk c resident in buffer p for all waves

    // kick off DMA for next chunk into the other buffer
    const int cn = c + gridDim.x;
    if (wave == 0 && cn < NCHUNK) {
      const int k0n = cn * KC;
      const uint32_t xb = xs_off + (uint32_t)(p ^ 1) * XBUF;
      const uint32_t wb_ = ws_off + (uint32_t)(p ^ 1) * WBUF;
      tdm_load(xb, x + k0n, XW0, GRU_IN, GRU_T, KC, GRU_T, GRU_IN);
      tdm_load(wb_ + 0 * KC * 64, Wr + (size_t)k0n * 16, WW0,
               (GRU_IN + 16) * 16, 1, KC * 16, 1, KC * 16);
      tdm_load(wb_ + 1 * KC * 64, Wz + (size_t)k0n * 16, WW0,
               (GRU_IN + 16) * 16, 1, KC * 16, 1, KC * 16);
      tdm_load(wb_ + 2 * KC * 64, Wh + (size_t)k0n * 16, WW0,
               (GRU_IN + 16) * 16, 1, KC * 16, 1, KC * 16);
    }

    // A: 16x4 fp32 (lane<16: K=k,k+1 ; lane>=16: K=k+2,k+3), B: 4x16 fp32 mirrored
    const float* xr = &xs[p][(mt * 16 + lmod) * XST + 2 * lhalf];
    const float* wb = &ws[p][nt * KC * 16 + (2 * lhalf) * 16 + lmod];
#pragma unroll
    for (int k = 0; k < KC; k += 8) {
      v2f a0; a0.x = xr[k];             a0.y = xr[k + 1];
      v2f b0; b0.x = wb[k * 16];        b0.y = wb[k * 16 + 16];
      acc0 = __builtin_amdgcn_wmma_f32_16x16x4_f32(false, a0, false, b0,
                                                   (short)0, acc0, false, false);
      v2f a1; a1.x = xr[k + 4];         a1.y = xr[k + 5];
      v2f b1; b1.x = wb[(k + 4) * 16];  b1.y = wb[(k + 4) * 16 + 16];
      acc1 = __builtin_amdgcn_wmma_f32_16x16x4_f32(false, a1, false, b1,
                                                   (short)0, acc1, false, false);
    }
    __syncthreads();   // compute on buffer p done before it is DMA'd again
    p ^= 1;
  }

  // flush: C layout -> t = mt*16 + vgpr + 8*(lane>=16), j = lane&15
  float* Gs = G + (blockIdx.x & (NSTRIPE - 1)) * GSTRIDE;
#pragma unroll
  for (int v = 0; v < 8; ++v) {
    const int t = mt * 16 + v + 8 * lhalf;
    if (t < GRU_T)
      atomicAdd(&Gs[nt * 512 + t * 16 + lmod], acc0[v] + acc1[v]);
  }
}

// ---------------- Phase B: 30-step recurrence on one wave ----------------
__global__ __launch_bounds__(32) void gru_tail(
    const float* __restrict__ G,
    const float* __restrict__ Ur, const float* __restrict__ br,
    const float* __restrict__ Uz, const float* __restrict__ bz,
    const float* __restrict__ Uh, const float* __restrict__ bh,
    const float* __restrict__ Wo, const float* __restrict__ bo,
    const float* __restrict__ Wf, const float* __restrict__ bf,
    float* __restrict__ out)
{
  __shared__ float U[3][16][16];
  __shared__ float Gx[3][GRU_T][16];
  __shared__ float hs[16];
  __shared__ float zr[16];
  const int lane = threadIdx.x;

  for (int i = lane; i < 256; i += 32) {
    U[0][i >> 4][i & 15] = Ur[i];
    U[1][i >> 4][i & 15] = Uz[i];
    U[2][i >> 4][i & 15] = Uh[i];
  }
  for (int i = lane; i < 3 * GRU_T * 16; i += 32) {  // reduce atomic stripes
    const int g = i / (GRU_T * 16), rem = i % (GRU_T * 16);
    float s = 0.f;
    for (int st = 0; st < NSTRIPE; ++st) s += G[st * GSTRIDE + g * 512 + rem];
    Gx[g][rem / 16][rem % 16] = s;
  }
  __syncthreads();

  const float brv = (lane < 16) ? br[lane] : 0.f;
  const float bzv = (lane < 16) ? bz[lane] : 0.f;
  const float bhv = (lane < 16) ? bh[lane] : 0.f;
  const float bov = (lane < 5)  ? bo[lane] : 0.f;

  float h = 0.f, acc = 0.f, z = 0.f;
  for (int t = 0; t < GRU_T; ++t) {
    __syncthreads();
    if (lane < 16) hs[lane] = h;
    __syncthreads();
    if (lane < 16) {
      float sr = Gx[0][t][lane] + brv;
      float sz = Gx[1][t][lane] + bzv;
      for (int i = 0; i < 16; ++i) {
        sr += hs[i] * U[0][i][lane];
        sz += hs[i] * U[1][i][lane];
      }
      const float r = 1.f / (1.f + __expf(-sr));
      z = 1.f / (1.f + __expf(-sz));
      zr[lane] = z * r;                 // nonstandard GRU: candidate uses z*r
    }
    __syncthreads();
    if (lane < 16) {
      float sh = Gx[2][t][lane] + bhv;
      for (int i = 0; i < 16; ++i) sh += zr[i] * U[2][i][lane];
      const float h1 = tanhf(sh);
      h = (1.f - z) * h + z * h1;
      hs[lane] = h;
    }
    __syncthreads();
    if (lane < 5) {
      float s = bov;
      for (int j = 0; j < 16; ++j) s += hs[j] * Wo[j * 5 + lane];
      acc += (1.f / (1.f + __expf(-s))) * Wf[t * 5 + lane];
    }
  }
  for (int off = 16; off > 0; off >>= 1) acc += __shfl_down(acc, off);
  if (lane == 0) out[0] = 1.f / (1.f + __expf(-(acc + bf[0])));
}

extern "C" void kernel_launch(void* const* d_in, const int* in_sizes, int n_in,
                              void* d_out, int out_size, void* d_ws, size_t ws_size,
                              hipStream_t stream) {
  const float* x  = (const float*)d_in[0];
  const float* Wr = (const float*)d_in[1];
  const float* br = (const float*)d_in[2];
  const float* Wz = (const float*)d_in[3];
  const float* bz = (const float*)d_in[4];
  const float* Wh = (const float*)d_in[5];
  const float* bh = (const float*)d_in[6];
  const float* Wo = (const float*)d_in[7];
  const float* bo = (const float*)d_in[8];
  const float* Wf = (const float*)d_in[9];
  const float* bf = (const float*)d_in[10];
  float* Gws = (float*)d_ws;

  hipMemsetAsync(d_ws, 0, (size_t)NSTRIPE * GSTRIDE * sizeof(float), stream);
  gru_gemm_wmma<<<NBLOCK, 192, 0, stream>>>(x, Wr, Wz, Wh, Gws);
  gru_tail<<<1, 32, 0, stream>>>(Gws,
                                 Wr + (size_t)GRU_IN * 16, br,
                                 Wz + (size_t)GRU_IN * 16, bz,
                                 Wh + (size_t)GRU_IN * 16, bh,
                                 Wo, bo, Wf, bf, (float*)d_out);
}